// GrNetwork_90718299226308
// MI455X (gfx1250) — compile-verified
//
#include <hip/hip_runtime.h>
#include <hip/hip_bf16.h>
#include <math.h>

// ---------------------------------------------------------------------------
// GrNetwork fused kernel for MI455X (gfx1250, wave32).
//
// Per batch element b (one wave32 workgroup each):
//   1. input[b] (400x10 f32) -> LDS, zero-padded to 400x16 (WMMA B panel).
//   2. X1 (96x10) = Wflat(96x400) @ input[b] via V_WMMA_F32_16X16X4_F32:
//      6 M-tiles x 100 K-steps. W streamed from global (L2-resident, 154 KB).
//   3. Per f: MGS orthonormalization of 12x10 X1[f] (lane = matrix row,
//      wave shuffle reductions); accumulate X4 += Q Q^T into LDS.
//      (Q Q^T is the colspace projector -> identical to Householder-QR result.)
//   4. Jacobi eigensolve of 12x12 X4 in LDS (8 cyclic sweeps), eigvecs in Vs.
//   5. X6 = I - v1 v1^T - v2 v2^T (bottom-2 eigvecs) fused directly into the
//      144x10 FC matvec (lane j computes logit j), then log_softmax -> out.
//
// HBM traffic = one pass over input (65.5 MB) + 164 KB output: ~2.8 us floor.
// ---------------------------------------------------------------------------

typedef float v2f __attribute__((ext_vector_type(2)));
typedef float v8f __attribute__((ext_vector_type(8)));

#define DIN   400
#define QD    10
#define DOUT  12
#define NF    8
#define M96   96      // NF * DOUT
#define NPAD  16

__device__ __forceinline__ float wave_sum(float v) {
#pragma unroll
    for (int off = 16; off; off >>= 1) v += __shfl_xor(v, off, 32);
    return v;
}

__global__ __launch_bounds__(32) void grnet_fused(
    const float* __restrict__ input,   // (B, 400, 10)
    const float* __restrict__ W,       // (8, 12, 400) == (96, 400) row-major
    const float* __restrict__ fc_w1,   // (144, 10)
    const float* __restrict__ fc_b1,   // (1, 10)
    float* __restrict__ out)           // (B, 10)
{
    __shared__ float Bs[DIN * NPAD];   // 25.6 KB: input[b] padded 400x16
    __shared__ float X1s[M96 * NPAD];  //  6.1 KB: X1 (96 rows x 16, 10 used)
    __shared__ float As[DOUT * DOUT];  // X4 accumulator -> Jacobi working mat
    __shared__ float Vs[DOUT * DOUT];  // Jacobi eigenvectors

    const int b    = blockIdx.x;
    const int lane = threadIdx.x;      // wave32: 0..31
    const int half = lane >> 4;        // 0 for lanes 0-15, 1 for 16-31
    const int lo   = lane & 15;

    const float* inb = input + (size_t)b * (DIN * QD);

    // ---- Stage 1: input[b] -> LDS B panel (zero-padded cols 10..15) ------
    for (int i = lane; i < DIN * NPAD; i += 32) Bs[i] = 0.0f;
    __syncthreads();
    for (int i = lane; i < DIN * QD; i += 32) {
        int r = i / QD, c = i - r * QD;
        Bs[r * NPAD + c] = inb[i];
    }
    __syncthreads();

    // ---- Stage 2: X1 = Wflat @ input[b] via WMMA f32 16x16x4 -------------
    // A (16x4 f32): lanes 0-15 -> K=k0,k0+1 ; lanes 16-31 -> K=k0+2,k0+3; M=lo
    // B (4x16 f32): V0 -> rows k0 / k0+2 ; V1 -> rows k0+1 / k0+3; N=lo
    v8f acc[6];
#pragma unroll
    for (int m = 0; m < 6; ++m) acc[m] = (v8f){0,0,0,0,0,0,0,0};

    for (int kk = 0; kk < DIN / 4; ++kk) {
        const int k0   = kk * 4;
        const int brow = k0 + 2 * half;
        v2f bv;
        bv.x = Bs[brow * NPAD + lo];
        bv.y = Bs[(brow + 1) * NPAD + lo];
#pragma unroll
        for (int m = 0; m < 6; ++m) {
            const float* ap = W + (size_t)(16 * m + lo) * DIN + k0 + 2 * half;
            v2f av;
            av.x = ap[0];
            av.y = ap[1];
            acc[m] = __builtin_amdgcn_wmma_f32_16x16x4_f32(
                false, av, false, bv, (short)0, acc[m], false, false);
        }
    }
    // C/D layout: VGPR v, lanes0-15 -> M=v, lanes16-31 -> M=v+8; N=lo
#pragma unroll
    for (int m = 0; m < 6; ++m) {
#pragma unroll
        for (int v = 0; v < 8; ++v) {
            int row = 16 * m + v + 8 * half;   // 0..95
            X1s[row * NPAD + lo] = acc[m][v];
        }
    }
    __syncthreads();

    // ---- Stage 3: per-f MGS + X4 accumulation ----------------------------
    for (int i = lane; i < DOUT * DOUT; i += 32) As[i] = 0.0f;
    __syncthreads();

    for (int f = 0; f < NF; ++f) {
        float x[QD];
#pragma unroll
        for (int q = 0; q < QD; ++q)
            x[q] = (lane < DOUT) ? X1s[(f * DOUT + lane) * NPAD + q] : 0.0f;

        // Modified Gram-Schmidt (all lanes run reductions; lanes>=12 hold 0)
#pragma unroll
        for (int j = 0; j < QD; ++j) {
            float nr  = wave_sum(x[j] * x[j]);
            float rin = rsqrtf(nr);
            x[j] *= rin;
#pragma unroll
            for (int k = j + 1; k < QD; ++k) {
                float d = wave_sum(x[j] * x[k]);
                x[k] = fmaf(-d, x[j], x[k]);
            }
        }
        // X4[lane][p] += dot(row_lane, row_p)
        for (int p = 0; p < DOUT; ++p) {
            float s = 0.0f;
#pragma unroll
            for (int q = 0; q < QD; ++q)
                s = fmaf(x[q], __shfl(x[q], p, 32), s);
            if (lane < DOUT) As[lane * DOUT + p] += s;
        }
        __syncthreads();
    }

    // ---- Stage 4: Jacobi eigensolve of As (12x12), eigvecs into Vs -------
    for (int i = lane; i < DOUT * DOUT; i += 32) Vs[i] = 0.0f;
    __syncthreads();
    if (lane < DOUT) Vs[lane * DOUT + lane] = 1.0f;
    __syncthreads();

    for (int sweep = 0; sweep < 8; ++sweep) {
        for (int p = 0; p < DOUT - 1; ++p) {
            for (int q = p + 1; q < DOUT; ++q) {
                float apq = As[p * DOUT + q];      // uniform broadcast reads
                float app = As[p * DOUT + p];
                float aqq = As[q * DOUT + q];
                float c, s;
                if (fabsf(apq) > 1e-20f) {
                    float tau = (aqq - app) / (2.0f * apq);
                    float t   = copysignf(1.0f, tau) /
                                (fabsf(tau) + sqrtf(1.0f + tau * tau));
                    c = rsqrtf(1.0f + t * t);
                    s = t * c;
                } else { c = 1.0f; s = 0.0f; }
                // A <- (A G): rotate columns p,q
                if (lane < DOUT) {
                    int i = lane;
                    float aip = As[i * DOUT + p], aiq = As[i * DOUT + q];
                    As[i * DOUT + p] = c * aip - s * aiq;
                    As[i * DOUT + q] = s * aip + c * aiq;
                }
                __syncthreads();
                // A <- G^T (A G): rotate rows p,q ; V <- V G
                if (lane < DOUT) {
                    int i = lane;
                    float api = As[p * DOUT + i], aqi = As[q * DOUT + i];
                    As[p * DOUT + i] = c * api - s * aqi;
                    As[q * DOUT + i] = s * api + c * aqi;
                    float vip = Vs[i * DOUT + p], viq = Vs[i * DOUT + q];
                    Vs[i * DOUT + p] = c * vip - s * viq;
                    Vs[i * DOUT + q] = s * vip + c * viq;
                }
                __syncthreads();
            }
        }
    }

    // ---- Stage 5: bottom-2 eigvecs -> X6 -> FC -> log_softmax ------------
    int   i1 = 0, i2 = 1;
    float e1 = INFINITY, e2 = INFINITY;
    for (int i = 0; i < DOUT; ++i) {            // uniform scan of diagonal
        float di = As[i * DOUT + i];
        if (di < e1)      { e2 = e1; i2 = i1; e1 = di; i1 = i; }
        else if (di < e2) { e2 = di; i2 = i; }
    }

    const int j = (lane < QD) ? lane : 0;       // lane j computes logit j
    float lg = fc_b1[j];
    for (int o = 0; o < DOUT; ++o) {
        float vo1 = Vs[o * DOUT + i1], vo2 = Vs[o * DOUT + i2];
        for (int p = 0; p < DOUT; ++p) {
            float x6 = ((o == p) ? 1.0f : 0.0f)
                     - vo1 * Vs[p * DOUT + i1]
                     - vo2 * Vs[p * DOUT + i2];
            lg = fmaf(x6, fc_w1[(o * DOUT + p) * QD + j], lg);
        }
    }
    // log_softmax across lanes 0..9
    float mx = (lane < QD) ? lg : -INFINITY;
#pragma unroll
    for (int off = 16; off; off >>= 1) mx = fmaxf(mx, __shfl_xor(mx, off, 32));
    float ex  = (lane < QD) ? expf(lg - mx) : 0.0f;
    float sum = wave_sum(ex);
    if (lane < QD) out[(size_t)b * QD + lane] = lg - mx - logf(sum);
}

extern "C" void kernel_launch(void* const* d_in, const int* in_sizes, int n_in,
                              void* d_out, int out_size, void* d_ws, size_t ws_size,
                              hipStream_t stream) {
    const float* input = (const float*)d_in[0];   // (B, 400, 10) f32
    const float* W     = (const float*)d_in[1];   // (8, 12, 400) f32
    const float* fc_w1 = (const float*)d_in[2];   // (144, 10) f32
    const float* fc_b1 = (const float*)d_in[3];   // (1, 10) f32
    float* out = (float*)d_out;                   // (B, 10) f32

    const int B = in_sizes[0] / (DIN * QD);       // 4096
    hipLaunchKernelGGL(grnet_fused, dim3(B), dim3(32), 0, stream,
                       input, W, fc_w1, fc_b1, out);
}